// gaussian_based_loss_7876970021273
// MI455X (gfx1250) — compile-verified
//
#include <hip/hip_runtime.h>
#include <math.h>

typedef __attribute__((ext_vector_type(2))) float v2f;
typedef __attribute__((ext_vector_type(8))) float v8f;

namespace {
constexpr int   Pn  = 1024;
constexpr int   Gn  = 1024;
constexpr int   NSI = 20;
constexpr float EPSf = 1e-8f;
constexpr float LN2  = 0.6931471805599453f;

// 2x2 row-major matmul: c = a @ b
__device__ __forceinline__ void mm2(const float a[4], const float b[4], float c[4]) {
  c[0] = fmaf(a[0], b[0], a[1] * b[2]);
  c[1] = fmaf(a[0], b[1], a[1] * b[3]);
  c[2] = fmaf(a[2], b[0], a[3] * b[2]);
  c[3] = fmaf(a[2], b[1], a[3] * b[3]);
}
} // namespace

__global__ __launch_bounds__(256) void gwd_loss_kernel(
    const float* __restrict__ pred_mu,    // P x 2
    const float* __restrict__ pred_conv,  // P x 2 x 2
    const float* __restrict__ gt_mu,      // G x 2
    const float* __restrict__ gt_conv,    // G x 2 x 2
    float* __restrict__ out)              // P x G
{
  const int lane = threadIdx.x & 31;
  const int wave = threadIdx.x >> 5;
  const int tile = blockIdx.x * 8 + wave;          // 4096 tiles total
  const int tiles_per_row = Gn / 16;               // 64
  const int i0 = (tile / tiles_per_row) * 16;
  const int j0 = (tile % tiles_per_row) * 16;

  const int half = lane >> 4;                      // 0: lanes 0-15, 1: lanes 16-31
  const int l15  = lane & 15;

  // ---- item1 tile via V_WMMA_F32_16X16X4_F32 ----
  // A[i,:] = [px, py, |p|^2, 1]  (16x4). Lanes 0-15 -> K=0,1 (M=lane);
  // lanes 16-31 -> K=2,3 (M=lane-16).
  const int ia = i0 + l15;
  const float px = pred_mu[2 * ia + 0];
  const float py = pred_mu[2 * ia + 1];
  v2f a;
  a.x = half ? fmaf(px, px, py * py) : px;
  a.y = half ? 1.0f : py;

  // B[:,j] = [-2gx, -2gy, 1, |g|^2]  (4x16). Lanes 0-15 -> K=0,1 (N=lane);
  // lanes 16-31 -> K=2,3 (N=lane-16).
  const int jb = j0 + l15;
  const float gx = gt_mu[2 * jb + 0];
  const float gy = gt_mu[2 * jb + 1];
  v2f b;
  b.x = half ? 1.0f : (-2.0f * gx);
  b.y = half ? fmaf(gx, gx, gy * gy) : (-2.0f * gy);

  v8f c = {};
  // item1[i0+r+8*half, j0+l15] lands in c[r] per the 16x16 f32 C/D layout.
  c = __builtin_amdgcn_wmma_f32_16x16x4_f32(
      /*neg_a=*/false, a, /*neg_b=*/false, b,
      /*c_mod=*/(short)0, c, /*reuse_a=*/false, /*reuse_b=*/false);

  // ---- per-lane column j is fixed: S2 = gt_conv[j], Q = S2^2 computed once ----
  const int j = j0 + l15;
  const float s2[4] = {gt_conv[4 * j + 0], gt_conv[4 * j + 1],
                       gt_conv[4 * j + 2], gt_conv[4 * j + 3]};
  float Q[4];
  mm2(s2, s2, Q);
  const float tr2 = Q[0] + Q[3];                   // tr(S2^2)

  float results[8];

#pragma unroll
  for (int r = 0; r < 8; ++r) {
    const int i = i0 + r + 8 * half;
    const float s1[4] = {pred_conv[4 * i + 0], pred_conv[4 * i + 1],
                         pred_conv[4 * i + 2], pred_conv[4 * i + 3]};
    // tr(S1^2)
    const float tr1 = fmaf(s1[0], s1[0], fmaf(2.0f * s1[1], s1[2], s1[3] * s1[3]));

    // M = S1 @ Q @ S1
    float U[4], M[4];
    mm2(s1, Q, U);
    mm2(U, s1, M);

    // Newton-Schulz sqrt of M (20 iters), all in registers.
    // Y = M / ||M||_F  via v_rsq_f32 (avoids the IEEE divide expansion).
    const float fro2 = fmaf(M[0], M[0], fmaf(M[1], M[1], fmaf(M[2], M[2], M[3] * M[3])));
    const float inv = __builtin_amdgcn_rsqf(fro2);             // 1/sqrt(fro2) = 1/||M||_F
    // sqrt(||M||_F) = fro2^(1/4), via two v_sqrt_f32
    const float sqrt_normA = __builtin_amdgcn_sqrtf(__builtin_amdgcn_sqrtf(fro2));

    float Y[4] = {M[0] * inv, M[1] * inv, M[2] * inv, M[3] * inv};
    float Z[4] = {1.0f, 0.0f, 0.0f, 1.0f};
#pragma unroll
    for (int it = 0; it < NSI; ++it) {
      float ZY[4], T[4], Yn[4], Zn[4];
      mm2(Z, Y, ZY);
      T[0] = 1.5f - 0.5f * ZY[0];
      T[1] = -0.5f * ZY[1];
      T[2] = -0.5f * ZY[2];
      T[3] = 1.5f - 0.5f * ZY[3];
      mm2(Y, T, Yn);
      mm2(T, Z, Zn);
      Y[0] = Yn[0]; Y[1] = Yn[1]; Y[2] = Yn[2]; Y[3] = Yn[3];
      Z[0] = Zn[0]; Z[1] = Zn[1]; Z[2] = Zn[2]; Z[3] = Zn[3];
    }
    // tr(sqrtM) = sqrt(||M||_F) * (Y00 + Y11)
    const float tr_sqrtM = sqrt_normA * (Y[0] + Y[3]);

    const float item2 = tr1 + tr2 - 2.0f * tr_sqrtM;
    const float wd = fmaxf(c[r] + item2, 0.0f);
    // log(sqrt(wd+eps)+1) via v_sqrt_f32 + v_log_f32 (log2) * ln2
    const float s = __builtin_amdgcn_sqrtf(wd + EPSf);
    results[r] = __builtin_amdgcn_logf(s + 1.0f) * LN2;
  }

#pragma unroll
  for (int r = 0; r < 8; ++r) {
    const int i = i0 + r + 8 * half;
    out[i * Gn + j] = results[r];
  }
}

extern "C" void kernel_launch(void* const* d_in, const int* in_sizes, int n_in,
                              void* d_out, int out_size, void* d_ws, size_t ws_size,
                              hipStream_t stream) {
  (void)in_sizes; (void)n_in; (void)d_ws; (void)ws_size; (void)out_size;
  const float* pred_mu   = (const float*)d_in[0];
  const float* pred_conv = (const float*)d_in[1];
  const float* gt_mu     = (const float*)d_in[2];
  const float* gt_conv   = (const float*)d_in[3];
  float* out = (float*)d_out;

  // 4096 tiles (64x64 tiles of 16x16), 8 waves (tiles) per 256-thread block.
  const int tiles = (Pn / 16) * (Gn / 16);
  const int blocks = tiles / 8;  // 512
  gwd_loss_kernel<<<blocks, 256, 0, stream>>>(pred_mu, pred_conv, gt_mu, gt_conv, out);
}